// MessagePassingLayer_88776974008405
// MI455X (gfx1250) — compile-verified
//
#include <hip/hip_runtime.h>

typedef __attribute__((ext_vector_type(2))) float v2f;
typedef __attribute__((ext_vector_type(8))) float v8f;

#define H 128
#define WAVES_PER_BLOCK 4

// ---------------------------------------------------------------------------
// Two-layer MLP over node rows using V_WMMA_F32_16X16X4_F32.
//   Y[m, :] = relu(X[m,:] @ W1^T + b1) @ W2^T + b2
// One wave computes a 16-row tile. K=128 -> 32 wmma steps per 16x16 tile,
// 8 column tiles per layer -> 512 wmma per wave.
// ---------------------------------------------------------------------------
__global__ __launch_bounds__(WAVES_PER_BLOCK * 32)
void mlp2_kernel(const float* __restrict__ X,
                 const float* __restrict__ W1, const float* __restrict__ B1,
                 const float* __restrict__ W2, const float* __restrict__ B2,
                 float* __restrict__ Y, int N)
{
    // Per-wave staging of the 16x128 hidden tile (cross-lane transpose between
    // the WMMA D layout and the A-fragment layout). Padded to dodge bank
    // conflicts: row stride 132 floats.
    __shared__ float hidden[WAVES_PER_BLOCK][16][H + 4];

    const int wave = threadIdx.x >> 5;
    const int lane = threadIdx.x & 31;
    const int lo   = lane & 15;      // M index (A/D) or N index (B/D)
    const int hi   = lane >> 4;      // selects K pair / upper rows

    const int row0 = (blockIdx.x * WAVES_PER_BLOCK + wave) * 16;
    if (row0 >= N) return;           // wave-uniform: EXEC stays all-ones

    // ---------------- Phase 1: hidden = relu(X @ W1^T + b1) ----------------
    v8f acc[8] = {};
    const float* __restrict__ arow = X + (size_t)(row0 + lo) * H;

    for (int kk = 0; kk < 32; ++kk) {
        const int k = kk * 4 + hi * 2;
        v2f a;
        a.x = arow[k];
        a.y = arow[k + 1];
#pragma unroll
        for (int t = 0; t < 8; ++t) {
            const int n = t * 16 + lo;               // output feature
            v2f b;                                   // B[k, n] = W1[n, k]
            b.x = W1[(size_t)n * H + k];
            b.y = W1[(size_t)n * H + k + 1];
            acc[t] = __builtin_amdgcn_wmma_f32_16x16x4_f32(
                false, a, false, b, (short)0, acc[t], false, false);
        }
    }

    // Bias + ReLU, stage to LDS. D layout: acc[t][r] at lane l is
    // C[M = r + 8*hi, N = t*16 + lo].
#pragma unroll
    for (int t = 0; t < 8; ++t) {
        const float bias = B1[t * 16 + lo];
#pragma unroll
        for (int r = 0; r < 8; ++r) {
            float h = acc[t][r] + bias;
            h = fmaxf(h, 0.0f);
            hidden[wave][r + 8 * hi][t * 16 + lo] = h;
        }
    }

    // Wave-local LDS RAW: DS ops from one wave are processed in order; the
    // explicit wait + memory clobber keeps the compiler from reordering.
    asm volatile("s_wait_dscnt 0" ::: "memory");

    // ---------------- Phase 2: Y = hidden @ W2^T + b2 ----------------------
    v8f acc2[8] = {};
    for (int kk = 0; kk < 32; ++kk) {
        const int k = kk * 4 + hi * 2;
        v2f a;
        a.x = hidden[wave][lo][k];
        a.y = hidden[wave][lo][k + 1];
#pragma unroll
        for (int t = 0; t < 8; ++t) {
            const int n = t * 16 + lo;
            v2f b;
            b.x = W2[(size_t)n * H + k];
            b.y = W2[(size_t)n * H + k + 1];
            acc2[t] = __builtin_amdgcn_wmma_f32_16x16x4_f32(
                false, a, false, b, (short)0, acc2[t], false, false);
        }
    }

#pragma unroll
    for (int t = 0; t < 8; ++t) {
        const float bias = B2[t * 16 + lo];
#pragma unroll
        for (int r = 0; r < 8; ++r) {
            const int m = row0 + r + 8 * hi;
            Y[(size_t)m * H + t * 16 + lo] = acc2[t][r] + bias;
        }
    }
}

// ---------------------------------------------------------------------------
// Edge scatter: out[dst] += msg[src] * w_e.
// One wave per edge, each lane moves a float4 (32 lanes * 4 = 128 channels).
// unsafeAtomicAdd -> global_atomic_add_f32 (no return, STOREcnt path).
// ---------------------------------------------------------------------------
__global__ __launch_bounds__(256)
void edge_scatter_kernel(const float* __restrict__ msg,
                         const long long* __restrict__ src_idx,
                         const long long* __restrict__ dst_idx,
                         const float* __restrict__ ew,
                         float* __restrict__ out, int E)
{
    const int lane   = threadIdx.x & 31;
    const int gwave  = (int)((blockIdx.x * (size_t)blockDim.x + threadIdx.x) >> 5);
    const int nwaves = (int)((gridDim.x * (size_t)blockDim.x) >> 5);

    for (int e = gwave; e < E; e += nwaves) {
        const long long s = src_idx[e];
        const long long d = dst_idx[e];
        const float w = ew[e];

        const float4* __restrict__ mp =
            reinterpret_cast<const float4*>(msg + (size_t)s * H);
        float4 v = mp[lane];

        float* __restrict__ op = out + (size_t)d * H + lane * 4;
        unsafeAtomicAdd(op + 0, v.x * w);
        unsafeAtomicAdd(op + 1, v.y * w);
        unsafeAtomicAdd(op + 2, v.z * w);
        unsafeAtomicAdd(op + 3, v.w * w);
    }
}

extern "C" void kernel_launch(void* const* d_in, const int* in_sizes, int n_in,
                              void* d_out, int out_size, void* d_ws, size_t ws_size,
                              hipStream_t stream)
{
    const float*     x   = (const float*)d_in[0];
    const long long* ei  = (const long long*)d_in[1];   // [2, E] int64
    const float*     ew  = (const float*)d_in[2];
    const float*     mw1 = (const float*)d_in[3];
    const float*     mb1 = (const float*)d_in[4];
    const float*     mw2 = (const float*)d_in[5];
    const float*     mb2 = (const float*)d_in[6];
    const float*     sw1 = (const float*)d_in[7];
    const float*     sb1 = (const float*)d_in[8];
    const float*     sw2 = (const float*)d_in[9];
    const float*     sb2 = (const float*)d_in[10];

    float* out    = (float*)d_out;
    float* msgbuf = (float*)d_ws;      // [N, H] fp32 message-MLP output

    const int N = in_sizes[0] / H;     // 50000
    const int E = in_sizes[2];         // 625000

    const int tiles  = (N + 15) / 16;
    const int blocks = (tiles + WAVES_PER_BLOCK - 1) / WAVES_PER_BLOCK;

    // self MLP -> d_out (initializes every output element)
    mlp2_kernel<<<blocks, WAVES_PER_BLOCK * 32, 0, stream>>>(
        x, sw1, sb1, sw2, sb2, out, N);
    // message MLP per NODE (MLP(x[src]) == MLP(x)[src]) -> workspace
    mlp2_kernel<<<blocks, WAVES_PER_BLOCK * 32, 0, stream>>>(
        x, mw1, mb1, mw2, mb2, msgbuf, N);
    // edge gather/scale/scatter-add
    edge_scatter_kernel<<<2048, 256, 0, stream>>>(
        msgbuf, ei, ei + E, ew, out, E);
}